// CTCLossLayer_48696339202439
// MI455X (gfx1250) — compile-verified
//
#include <hip/hip_runtime.h>

// CTC loss, B=64 T=1000 V=512 L=200 (S=401), fp32.
// Phase 2 first in file so the disasm snippet shows the TDM path.
//   - 8-slot LDS ring of logits rows, 7 TDM loads in flight (wave 0 drives),
//     s_wait_tensorcnt(6) => oldest row landed (TENSORcnt is in-order),
//     one workgroup barrier per timestep.
// Phase 1: denom[b,t] = logsumexp_V (bandwidth-bound, wave-per-row, shfl_xor)
// Phase 3: mean reduction to scalar.

#define CTC_B 64
#define CTC_T 1000
#define CTC_V 512
#define CTC_L 200
#define CTC_S 401          // 2L+1
#define CTC_BLANK 511
#define CTC_NEG (-1e30f)
#define CTC_NBUF 8         // ring slots; wait constant = NBUF-2 = 6
#define CTC_ROWB (CTC_V * 4)  // bytes per row

typedef __attribute__((ext_vector_type(4))) unsigned int v4u;
typedef __attribute__((ext_vector_type(8))) int          v8i;
typedef __attribute__((ext_vector_type(4))) int          v4i;

// ---------------------------------------------------------------------------
// TDM: DMA one 512-float logits row from global into LDS (1-D tile).
// Descriptor layout per CDNA5 ISA (groups 0/1; rest 0).
// ---------------------------------------------------------------------------
__device__ __forceinline__ void tdm_load_row(unsigned long long gaddr,
                                             unsigned lds_addr) {
  v4u g0;
  g0.x = 1u;                                           // count=1, user desc
  g0.y = lds_addr;                                     // LDS byte address
  g0.z = (unsigned)(gaddr & 0xffffffffull);            // global_addr[31:0]
  g0.w = (unsigned)((gaddr >> 32) & 0x01ffffffull)     // global_addr[56:32]
         | 0x80000000u;                                // type=2 ("image")
  v8i g1;
  g1[0] = 0x20000;                  // wg_mask=0, data_size=2 (4 bytes)
  g1[1] = (int)(CTC_V << 16);       // tensor_dim0[15:0] in bits[63:48] -> 512
  g1[2] = (int)(1 << 16);           // tensor_dim0 hi=0 | tensor_dim1 lo = 1
  g1[3] = (int)(CTC_V << 16);       // tensor_dim1 hi=0 | tile_dim0 = 512
  g1[4] = 0;                        // tile_dim1=0 (unused), tile_dim2=0
  g1[5] = CTC_V;                    // tensor_dim0_stride = 512 elements
  g1[6] = 0;                        // stride0 hi | stride1 lo
  g1[7] = 0;                        // stride1 hi
  v4i z4 = {0, 0, 0, 0};            // groups 2/3: dims beyond 2D unused
  v8i z8 = {0, 0, 0, 0, 0, 0, 0, 0};
  __builtin_amdgcn_tensor_load_to_lds(g0, g1, z4, z4, z8, /*cpol=*/0);
}

// ---------------------------------------------------------------------------
// Kernel 2: forward scan. grid=B, block=416 (13 waves, covers S=401).
// ---------------------------------------------------------------------------
__global__ __launch_bounds__(416)
void ctc_alpha_kernel(const float* __restrict__ logits,
                      const int* __restrict__ labels,
                      const float* __restrict__ denom,
                      float* __restrict__ loss) {
  const int b   = blockIdx.x;
  const int tid = threadIdx.x;

  __shared__ float rowbuf[CTC_NBUF][CTC_V];   // 16 KB ring of logits rows
  __shared__ float alpha[2][CTC_S + 1];
  __shared__ int   lab[CTC_S + 1];
  __shared__ int   skipm[CTC_S + 1];
  __shared__ int   llsh;

  const unsigned long long base =
      (unsigned long long)(uintptr_t)(logits + (size_t)b * CTC_T * CTC_V);
  const unsigned lds_base = (unsigned)(uintptr_t)(&rowbuf[0][0]);

  // Fill the pipeline: rows 0..NBUF-2 -> slots 0..NBUF-2 (7 in flight).
  if (tid < 32) {
#pragma unroll
    for (int r = 0; r < CTC_NBUF - 1; ++r)
      tdm_load_row(base + (unsigned long long)r * CTC_ROWB,
                   lds_base + (unsigned)r * CTC_ROWB);
  }

  // Extended label sequence + skip mask + label length (overlaps the DMAs).
  if (tid == 0) llsh = 0;
  if (tid < CTC_S)
    lab[tid] = (tid & 1) ? labels[b * CTC_L + (tid >> 1)] : CTC_BLANK;
  __syncthreads();
  if (tid < CTC_S) {
    int lv = lab[tid];
    int p2 = (tid >= 2) ? lab[tid - 2] : -1;
    skipm[tid] = (lv != CTC_BLANK && lv != p2) ? 1 : 0;
  }
  if (tid < CTC_L && labels[b * CTC_L + tid] != 0) atomicAdd(&llsh, 1);

  int p = 0;
  for (int t = 0; t < CTC_T; ++t) {
    // TENSORcnt completes in order: <=6 outstanding => row t has landed.
    if (tid < 32) __builtin_amdgcn_s_wait_tensorcnt(CTC_NBUF - 2);
    __syncthreads();  // publishes: TDM row t, alpha writes from step t-1

    // Keep 7 in flight: issue row t+7 (clamped) into slot (t+7)%8 = (t-1)%8,
    // whose only consumer was step t-1 (reads done before this barrier).
    if (tid < 32) {
      int r = (t + CTC_NBUF - 1 < CTC_T) ? (t + CTC_NBUF - 1) : (CTC_T - 1);
      tdm_load_row(base + (unsigned long long)r * CTC_ROWB,
                   lds_base + (unsigned)((t + CTC_NBUF - 1) & (CTC_NBUF - 1)) *
                                  CTC_ROWB);
    }

    const float dn = denom[b * CTC_T + t];
    const float* row = rowbuf[t & (CTC_NBUF - 1)];
    if (tid < CTC_S) {
      if (t == 0) {
        alpha[0][tid] = (tid < 2) ? (row[lab[tid]] - dn) : CTC_NEG;
      } else {
        float a0 = alpha[p][tid];
        float a1 = (tid >= 1) ? alpha[p][tid - 1] : CTC_NEG;
        float a2 = (tid >= 2 && skipm[tid]) ? alpha[p][tid - 2] : CTC_NEG;
        float m  = fmaxf(a0, fmaxf(a1, a2));
        float lse = m + logf(expf(a0 - m) + expf(a1 - m) + expf(a2 - m));
        alpha[p ^ 1][tid] = lse + (row[lab[tid]] - dn);
      }
    }
    if (t) p ^= 1;
  }
  __syncthreads();

  if (tid == 0) {
    int ll = llsh;
    int ib = min(2 * ll, CTC_S - 1);
    int il = max(min(2 * ll - 1, CTC_S - 1), 0);
    float ab = alpha[p][ib], al = alpha[p][il];
    float m = fmaxf(ab, al);
    loss[b] = -(m + logf(expf(ab - m) + expf(al - m)));
  }
}

// ---------------------------------------------------------------------------
// Kernel 1: denom[row] = logsumexp over V=512.  One wave32 per row.
// ---------------------------------------------------------------------------
__global__ __launch_bounds__(256)
void ctc_denom_kernel(const float* __restrict__ logits,
                      float* __restrict__ denom) {
  const int wave = threadIdx.x >> 5;
  const int lane = threadIdx.x & 31;
  const int row  = blockIdx.x * (blockDim.x >> 5) + wave;
  if (row >= CTC_B * CTC_T) return;

  const float4* rp = (const float4*)(logits + (size_t)row * CTC_V);
  float4 v[4];
#pragma unroll
  for (int j = 0; j < 4; ++j) v[j] = rp[j * 32 + lane];  // coalesced 16B/lane

  float m = -3.402823e38f;
#pragma unroll
  for (int j = 0; j < 4; ++j)
    m = fmaxf(m, fmaxf(fmaxf(v[j].x, v[j].y), fmaxf(v[j].z, v[j].w)));
  float s = 0.f;
#pragma unroll
  for (int j = 0; j < 4; ++j)
    s += expf(v[j].x - m) + expf(v[j].y - m) + expf(v[j].z - m) +
         expf(v[j].w - m);

  // wave32 butterfly combine of (max, sumexp)
#pragma unroll
  for (int d = 16; d; d >>= 1) {
    float m2 = __shfl_xor(m, d, 32);
    float s2 = __shfl_xor(s, d, 32);
    float M  = fmaxf(m, m2);
    s = s * expf(m - M) + s2 * expf(m2 - M);
    m = M;
  }
  if (lane == 0) denom[row] = m + logf(s);
}

// ---------------------------------------------------------------------------
// Kernel 3: scalar mean over 64 losses.
// ---------------------------------------------------------------------------
__global__ void ctc_final_kernel(const float* __restrict__ loss,
                                 float* __restrict__ out) {
  int lane = threadIdx.x;
  float v = loss[lane] + loss[lane + 32];
#pragma unroll
  for (int d = 16; d; d >>= 1) v += __shfl_xor(v, d, 32);
  if (lane == 0) out[0] = v * (1.0f / 64.0f) + 1e-7f;
}

extern "C" void kernel_launch(void* const* d_in, const int* in_sizes, int n_in,
                              void* d_out, int out_size, void* d_ws,
                              size_t ws_size, hipStream_t stream) {
  const float* y_pred = (const float*)d_in[0];  // (B,T,V) fp32
  const int*   y_true = (const int*)d_in[1];    // (B,L) int

  float* denom = (float*)d_ws;                  // B*T floats = 256 KB
  float* loss  = denom + CTC_B * CTC_T;         // B floats

  ctc_denom_kernel<<<(CTC_B * CTC_T) / 8, 256, 0, stream>>>(y_pred, denom);
  ctc_alpha_kernel<<<CTC_B, 416, 0, stream>>>(y_pred, y_true, denom, loss);
  ctc_final_kernel<<<1, 32, 0, stream>>>(loss, (float*)d_out);
}